// RTDLiteRegularizer_31396210934340
// MI455X (gfx1250) — compile-verified
//
#include <hip/hip_runtime.h>
#include <hip/hip_bf16.h>
#include <math.h>

// RTD-Lite regularizer for MI455X (gfx1250, wave32).
// Inputs (setup_inputs order): logits (16,2,512,512) f32, tour_edges (16,512,2) i64,
//                              dist (16,512,512) f32.  Output: 1 float.
#define NN 512
#define BB 16

typedef __attribute__((ext_vector_type(2))) float v2f;
typedef __attribute__((ext_vector_type(8))) float v8f;

// ---------------- workspace layout (floats) ----------------
//  [0]            : max(dist)
//  [1  .. 17)     : mstA[16]
//  [17 .. 33)     : mstC[16]
//  [64 .. 1088)   : max partials (1024)
//  [2048 .. 4096) : l1 block partials (2048)
//  [8192 .. )     : int nxt[8192], int prv[8192]
//  [32768 .. )    : optional materialized WC (16 MB) if ws_size allows

// ---- Kernel 1: per-block max over dist (float4 loads, deterministic) ----
__global__ void k_max_partial(const float* __restrict__ dist,
                              float* __restrict__ part) {
  __shared__ float sm[8];
  const float4* d4 = (const float4*)dist;
  float m = 0.0f;                                   // dist >= 0
  for (int i = blockIdx.x * 256 + threadIdx.x; i < (BB * NN * NN) / 4;
       i += 1024 * 256) {
    float4 v = d4[i];
    m = fmaxf(m, fmaxf(fmaxf(v.x, v.y), fmaxf(v.z, v.w)));
  }
  for (int off = 16; off; off >>= 1) m = fmaxf(m, __shfl_xor(m, off, 32));
  if ((threadIdx.x & 31) == 0) sm[threadIdx.x >> 5] = m;
  __syncthreads();
  if (threadIdx.x == 0) {
    float r = sm[0];
    for (int w = 1; w < 8; ++w) r = fmaxf(r, sm[w]);
    part[blockIdx.x] = r;
  }
}

// ---- Kernel 2: reduce 1024 partial maxes -> ws[0] ----
__global__ void k_max_final(const float* __restrict__ part,
                            float* __restrict__ outmax) {
  __shared__ float sm[256];
  float m = fmaxf(fmaxf(part[threadIdx.x], part[threadIdx.x + 256]),
                  fmaxf(part[threadIdx.x + 512], part[threadIdx.x + 768]));
  sm[threadIdx.x] = m;
  __syncthreads();
  for (int s = 128; s; s >>= 1) {
    if (threadIdx.x < s) sm[threadIdx.x] = fmaxf(sm[threadIdx.x], sm[threadIdx.x + s]);
    __syncthreads();
  }
  if (threadIdx.x == 0) outmax[0] = sm[0];
}

// ---- Kernel 3: tour-edge neighbor table (each vertex: next & prev in cycle) ----
__global__ void k_nbr(const long long* __restrict__ te,
                      int* __restrict__ nxt, int* __restrict__ prv) {
  int e = blockIdx.x * 256 + threadIdx.x;           // 8192 edges
  if (e < BB * NN) {
    int b = e / NN;
    int i = (int)te[2 * e];
    int j = (int)te[2 * e + 1];
    nxt[b * NN + i] = j;                            // permutation: no races
    prv[b * NN + j] = i;
  }
}

// ---- Kernel 4: l1 = sum |WA - WB| via WMMA f32 16x16x4 accumulation ----
// Ones-matrix trick: A = all-ones 16x4 => D = colsum(B) + C. Each iteration a
// lane contributes one f32 partial; one v_wmma folds the wave's 32 partials
// into the f32 accumulator tile. Also (optionally) materializes WC = min(WA,WB)
// so the Prim critical path reads one row instead of three + expf.
__global__ void k_l1(const float* __restrict__ logits,
                     const float* __restrict__ dist,
                     const int* __restrict__ nxt, const int* __restrict__ prv,
                     const float* __restrict__ maxv,
                     float* __restrict__ part,
                     float* __restrict__ wcout) {
  const float BIG = 10.0f * maxv[0];
  v8f c = {};
  v2f onesv; onesv[0] = 1.0f; onesv[1] = 1.0f;
  const int total4 = (BB * NN * NN) / 4;            // 1,048,576
  const int stride = 2048 * 256;                    // exact 2 iters/lane, EXEC full
  for (int i4 = blockIdx.x * 256 + threadIdx.x; i4 < total4; i4 += stride) {
    int base = i4 * 4;
    int j   = base & (NN - 1);
    int row = base >> 9;                            // b*NN + i
    int i   = row & (NN - 1);
    int b   = row >> 9;
    float4 d  = *(const float4*)(dist + (size_t)base);
    size_t lb = (((size_t)b * 2) * NN + i) * NN + j;
    float4 l0 = *(const float4*)(logits + lb);
    float4 l1 = *(const float4*)(logits + lb + (size_t)NN * NN);
    int nx = nxt[row];
    int pv = prv[row];
    const float* dp = (const float*)&d;
    const float* p0 = (const float*)&l0;
    const float* p1 = (const float*)&l1;
    float s = 0.0f;
    float4 wcv;
    float* wcp = (float*)&wcv;
#pragma unroll
    for (int t = 0; t < 4; ++t) {
      float dd = dp[t];
      float p  = 1.0f / (1.0f + expf(p0[t] - p1[t]));   // softmax ch-1
      float wb = (1.0f - p) * BIG + p * dd;
      int jj   = j + t;
      float wa = (jj == i) ? 0.0f : ((jj == nx || jj == pv) ? dd : BIG);
      s += fabsf(wa - wb);
      wcp[t] = fminf(wa, wb);
    }
    if (wcout) *(float4*)(wcout + (size_t)base) = wcv;  // uniform branch
    v2f bv; bv[0] = s; bv[1] = 0.0f;
    // (neg_a, A, neg_b, B, c_mod, C, reuse_a, reuse_b)
    c = __builtin_amdgcn_wmma_f32_16x16x4_f32(false, onesv, false, bv,
                                              (short)0, c, false, false);
  }
  // c[0]: rows 0/8 of D (all rows identical); column n holds lanes n and n+16
  // -> 32-lane sum = 2x wave total.
  float ps = c[0];
  for (int off = 16; off; off >>= 1) ps += __shfl_xor(ps, off, 32);
  ps *= 0.5f;
  __shared__ float wsum[8];
  if ((threadIdx.x & 31) == 0) wsum[threadIdx.x >> 5] = ps;
  __syncthreads();
  if (threadIdx.x == 0) {
    float r = 0.0f;
    for (int w = 0; w < 8; ++w) r += wsum[w];       // fixed order: deterministic
    part[blockIdx.x] = r;
  }
}

// ---- Kernel 5: key-array Prim, 32 blocks = 16 graphs x {WA, WC}. ----
// Keys + selected flag live in registers (thread == vertex). Per step:
// wave-level u64 packed argmin via xor-shuffles (5 steps), 16 wave leaders ->
// LDS, 1 barrier, 16-lane reduce in wave 0, 1 barrier. 2 barriers/step vs 10.
__global__ void __launch_bounds__(512)
k_prim(const float* __restrict__ logits, const float* __restrict__ dist,
       const int* __restrict__ nxt, const int* __restrict__ prv,
       const float* __restrict__ maxv, const float* __restrict__ wc,
       float* __restrict__ mstA, float* __restrict__ mstC) {
  const int  blk  = blockIdx.x;
  const bool isC  = blk >= BB;
  const int  b    = blk & (BB - 1);
  const int  tid  = threadIdx.x;                    // 0..511 == vertex id
  const int  wv   = tid >> 5;
  const int  lane = tid & 31;
  __shared__ unsigned long long wred[16];
  __shared__ int s_v;
  const float BIG = 10.0f * maxv[0];
  float key = INFINITY;                             // register key per vertex
  bool  sel = false;
  float tot = 0.0f;                                 // meaningful in (wave0,lane0)
  if (tid == 0) s_v = 0;
  __syncthreads();
  for (int step = 0; step < NN - 1; ++step) {
    const int v = s_v;
    if (tid == v) sel = true;                       // newly selected vertex
    // weight W(v, tid), generated on the fly
    float w;
    if (isC && wc) {
      w = wc[((size_t)b * NN + v) * NN + tid];      // L2-resident row
    } else {
      int nx = nxt[b * NN + v];
      int pv = prv[b * NN + v];
      float d  = dist[((size_t)b * NN + v) * NN + tid];
      float wa = (tid == nx || tid == pv) ? d : BIG;
      w = wa;
      if (isC) {                                    // fallback: WC on the fly
        size_t lb = (((size_t)b * 2) * NN + v) * NN + tid;
        float p  = 1.0f / (1.0f + expf(logits[lb] - logits[lb + (size_t)NN * NN]));
        w = fminf(wa, (1.0f - p) * BIG + p * d);
      }
    }
    if (!sel) key = fminf(key, w);
    // packed argmin: weights >= 0 so float bits are order-preserving;
    // low 32 bits = index -> min() gives first-min tie-break like reference.
    float cand = sel ? INFINITY : key;
    unsigned long long pk =
        ((unsigned long long)__float_as_uint(cand) << 32) | (unsigned int)tid;
    for (int m = 16; m; m >>= 1) {
      unsigned long long q = __shfl_xor(pk, m, 32);
      pk = (q < pk) ? q : pk;
    }
    if (lane == 0) wred[wv] = pk;
    __syncthreads();
    if (wv == 0) {
      unsigned long long p2 = wred[lane & 15];
      for (int m = 8; m; m >>= 1) {
        unsigned long long q = __shfl_xor(p2, m, 16);
        p2 = (q < p2) ? q : p2;
      }
      if (lane == 0) {
        s_v = (int)(unsigned int)(p2 & 0xffffffffu);
        tot += __uint_as_float((unsigned int)(p2 >> 32)); // selection order = ref
      }
    }
    __syncthreads();
  }
  if (tid == 0) (isC ? mstC : mstA)[b] = tot;
}

// ---- Kernel 6: finalize scalar loss ----
__global__ void k_final(const float* __restrict__ part,
                        const float* __restrict__ mstA,
                        const float* __restrict__ mstC,
                        float* __restrict__ out) {
  __shared__ float sm[256];
  float s = 0.0f;
  for (int i = threadIdx.x; i < 2048; i += 256) s += part[i];
  sm[threadIdx.x] = s;
  __syncthreads();
  for (int r = 128; r; r >>= 1) {
    if (threadIdx.x < r) sm[threadIdx.x] += sm[threadIdx.x + r];
    __syncthreads();
  }
  if (threadIdx.x == 0) {
    float l1 = sm[0] / (float)((size_t)BB * NN * NN);
    float topo = 0.0f;
    for (int g = 0; g < BB; ++g) topo += mstA[g] - mstC[g];
    topo = topo / (float)BB + 0.001f * l1;
    out[0] = 1.5e-5f * topo;                        // MAX_COEF * step / WARMUP
  }
}

extern "C" void kernel_launch(void* const* d_in, const int* in_sizes, int n_in,
                              void* d_out, int out_size, void* d_ws, size_t ws_size,
                              hipStream_t stream) {
  const float*     logits = (const float*)d_in[0];
  const long long* te     = (const long long*)d_in[1];
  const float*     dist   = (const float*)d_in[2];
  float* ws       = (float*)d_ws;
  float* w_max    = ws + 0;
  float* w_mstA   = ws + 1;
  float* w_mstC   = ws + 17;
  float* w_maxprt = ws + 64;
  float* w_l1prt  = ws + 2048;
  int*   w_nxt    = (int*)(ws + 8192);
  int*   w_prv    = (int*)(ws + 8192) + BB * NN;
  // Materialize WC (16 MB) if the scratch allocation allows; decision depends
  // only on ws_size, so work is identical on every call (graph-capture safe).
  const size_t WC_OFF = 32768;                      // floats
  const size_t WC_ELT = (size_t)BB * NN * NN;
  float* w_wc = (ws_size >= (WC_OFF + WC_ELT) * sizeof(float)) ? (ws + WC_OFF)
                                                               : nullptr;

  k_max_partial<<<1024, 256, 0, stream>>>(dist, w_maxprt);
  k_max_final  <<<1,    256, 0, stream>>>(w_maxprt, w_max);
  k_nbr        <<<32,   256, 0, stream>>>(te, w_nxt, w_prv);
  k_l1         <<<2048, 256, 0, stream>>>(logits, dist, w_nxt, w_prv, w_max,
                                          w_l1prt, w_wc);
  k_prim       <<<32,   512, 0, stream>>>(logits, dist, w_nxt, w_prv, w_max,
                                          w_wc, w_mstA, w_mstC);
  k_final      <<<1,    256, 0, stream>>>(w_l1prt, w_mstA, w_mstC, (float*)d_out);
}